// draw_83064667504790
// MI455X (gfx1250) — compile-verified
//
#include <hip/hip_runtime.h>
#include <math.h>

// ---------------------------------------------------------------------------
// DRAW-style VAE forward for MI455X (gfx1250).
// All GEMMs use V_WMMA_F32_16X16X4_F32 (full fp32 matrix op, wave32).
// ---------------------------------------------------------------------------

typedef float v2f __attribute__((ext_vector_type(2)));
typedef float v8f __attribute__((ext_vector_type(8)));

#define BM 64
#define BN 64
#define BK 32

// Generic fp32 WMMA GEMM:  C[M,N] (ldc) (+)= A[M,K](lda) * B[K,N](ldb) (+ bias)
// Requirements (guaranteed by padded buffers): M%64==0, N%64==0, K%32==0.
// Block = 512 threads = 16 waves; each wave owns one 16x16 tile of the 64x64
// block tile; 8 wmma(16x16x4) per 32-wide K panel staged through LDS.
__global__ __launch_bounds__(512)
void gemm_f32_wmma(const float* __restrict__ A, int lda,
                   const float* __restrict__ Bm, int ldb,
                   float* __restrict__ C, int ldc,
                   int K,
                   const float* __restrict__ bias,   // len >= N or nullptr
                   int accumulate)
{
    __shared__ float As[BK][BM + 1];   // [k][m], +1 pad kills bank conflicts
    __shared__ float Bs[BK][BN + 1];   // [k][n]

    const int tid  = threadIdx.x;          // 0..511
    const int wave = tid >> 5;             // 0..15
    const int lane = tid & 31;
    const int wm   = wave >> 2;            // 0..3 : sub-tile row
    const int wn   = wave & 3;             // 0..3 : sub-tile col
    const int l16  = lane & 15;
    const int lHi  = lane >> 4;            // 0/1 (K half select per ISA layout)

    const int mBlk = blockIdx.y * BM;
    const int nBlk = blockIdx.x * BN;
    const int cm   = mBlk + wm * 16;               // tile row base
    const int cn   = nBlk + wn * 16 + l16;         // this lane's output column

    // Hoisted epilogue operand: one guarded bias load (overlaps K-loop),
    // branch-free store loop below.
    float bv = 0.f;
    if (bias) bv = bias[cn];

    v8f acc = {};
    if (accumulate) {
#pragma unroll
        for (int r = 0; r < 8; ++r)
            acc[r] = C[(size_t)(cm + r + 8 * lHi) * ldc + cn];
    }

    // ---- staging pointers (strength-reduced; 4 independent loads/iter) ----
    // A: this thread covers rows (tid>>5)+{0,16,32,48} at fixed k = tid&31.
    // B: this thread covers k rows (tid>>6)+{0,8,16,24} at fixed n = tid&63.
    const int aRow = tid >> 5, aK = tid & 31;
    const int bK   = tid >> 6, bN = tid & 63;
    const float* ap = A  + (size_t)(mBlk + aRow) * lda + aK;
    const float* bp = Bm + (size_t)bK * ldb + (nBlk + bN);
    float* asd = &As[aK][aRow];
    float* bsd = &Bs[bK][bN];
    const size_t aStride = (size_t)16 * lda;   // 16 rows per sub-iteration
    const size_t bStride = (size_t)8 * ldb;    // 8 k-rows per sub-iteration

    for (int kb = 0; kb < K; kb += BK) {
#pragma unroll
        for (int i = 0; i < 4; ++i) {
            asd[i * 16]            = ap[(size_t)i * aStride];
            bsd[i * 8 * (BN + 1)]  = bp[(size_t)i * bStride];
        }
        ap += BK;                       // advance K window in A
        bp += (size_t)BK * ldb;         // advance K window in B
        __syncthreads();

#pragma unroll
        for (int kk = 0; kk < BK; kk += 4) {
            // A frag: lane l, reg r -> A[M = l%16][K = 2*(l/16)+r]
            // B frag: lane l, reg r -> B[K = 2*(l/16)+r][N = l%16]
            v2f af, bf;
            af[0] = As[kk + 2 * lHi + 0][wm * 16 + l16];
            af[1] = As[kk + 2 * lHi + 1][wm * 16 + l16];
            bf[0] = Bs[kk + 2 * lHi + 0][wn * 16 + l16];
            bf[1] = Bs[kk + 2 * lHi + 1][wn * 16 + l16];
            acc = __builtin_amdgcn_wmma_f32_16x16x4_f32(
                      false, af, false, bf, (short)0, acc, false, false);
        }
        __syncthreads();
    }

#pragma unroll
    for (int r = 0; r < 8; ++r)
        C[(size_t)(cm + r + 8 * lHi) * ldc + cn] = acc[r] + bv;
}

// ---------------------------------------------------------------------------
// Layout constants (all K/N slots padded to WMMA-friendly multiples)
// ENC row (width 1952): [0,784)=x  [784,1568)=xhat  [1568,1696)=h_dec|pad
//                       [1696,1824)=h_mu|pad        [1824,1952)=h_lv|pad
// GI  row (width 1280): [0,320)=gi_mu [320,640)=gi_lv
//                       [640,960)=gh_mu [960,1280)=gh_lv   (each r|z|n @ +0/+100/+200)
// ---------------------------------------------------------------------------
#define ENC_W 1952
#define GI_W  1280
#define PRJ_W 256
#define ZH_W  256
#define GD_W  640
#define C_W   832

__global__ void zero_f(float* p, long n) {
    long i = (long)blockIdx.x * blockDim.x + threadIdx.x;
    if (i < n) p[i] = 0.f;
}

__global__ void init_enc(const float* __restrict__ x, float* __restrict__ ENC, int B) {
    long i = (long)blockIdx.x * blockDim.x + threadIdx.x;
    if (i >= (long)B * ENC_W) return;
    int col = (int)(i % ENC_W);
    long row = i / ENC_W;
    ENC[i] = (col < 784) ? x[row * 784 + col] : 0.f;
}

// Wenc[1952][1280]
__global__ void pack_wenc(const float* Wih_mu, const float* Wih_lv,
                          const float* Whh_mu, const float* Whh_lv, float* Wp) {
    long i = (long)blockIdx.x * blockDim.x + threadIdx.x;
    if (i >= (long)ENC_W * GI_W) return;
    int k = (int)(i / GI_W), n = (int)(i % GI_W);
    float v = 0.f;
    if (n < 300 && k < 1668)                                  v = Wih_mu[(long)n * 1668 + k];
    else if (n >= 320 && n < 620 && k < 1668)                 v = Wih_lv[(long)(n - 320) * 1668 + k];
    else if (n >= 640 && n < 940 && k >= 1696 && k < 1796)    v = Whh_mu[(n - 640) * 100 + (k - 1696)];
    else if (n >= 960 && n < 1260 && k >= 1824 && k < 1924)   v = Whh_lv[(n - 960) * 100 + (k - 1824)];
    Wp[i] = v;
}

// Wprj[256][256] block-diag: rows 0..99 -> Wmu cols 0..99 ; rows 128..227 -> Wlv cols 128..227
__global__ void pack_wprj(const float* Wmu, const float* Wlv, float* Wp) {
    int i = blockIdx.x * blockDim.x + threadIdx.x;
    if (i >= 256 * 256) return;
    int k = i / 256, n = i % 256;
    float v = 0.f;
    if (n < 100 && k < 100)                               v = Wmu[n * 100 + k];
    else if (n >= 128 && n < 228 && k >= 128 && k < 228)  v = Wlv[(n - 128) * 100 + (k - 128)];
    Wp[i] = v;
}

// Wdec[256][640]: A = [z|pad|h_dec|pad]; cols [0,320)=gi_dec, [320,640)=gh_dec
__global__ void pack_wdec(const float* Wih_dec, const float* Whh_dec, float* Wp) {
    int i = blockIdx.x * blockDim.x + threadIdx.x;
    if (i >= 256 * 640) return;
    int k = i / 640, n = i % 640;
    float v = 0.f;
    if (n < 300 && k < 100)                               v = Wih_dec[n * 100 + k];
    else if (n >= 320 && n < 620 && k >= 128 && k < 228)  v = Whh_dec[(n - 320) * 100 + (k - 128)];
    Wp[i] = v;
}

// Wwr[128][832], bias_wr[832]
__global__ void pack_wwr(const float* Wwrite, float* Wp) {
    int i = blockIdx.x * blockDim.x + threadIdx.x;
    if (i >= 128 * C_W) return;
    int k = i / C_W, n = i % C_W;
    Wp[i] = (n < 784 && k < 100) ? Wwrite[n * 100 + k] : 0.f;
}
__global__ void pack_bwr(const float* bwrite, float* bp) {
    int i = blockIdx.x * blockDim.x + threadIdx.x;
    if (i < C_W) bp[i] = (i < 784) ? bwrite[i] : 0.f;
}

// xhat = x - sigmoid(c)  -> ENC[:,784:1568]
__global__ void k_xhat(const float* __restrict__ x, const float* __restrict__ c,
                       float* __restrict__ ENC, int B) {
    long i = (long)blockIdx.x * blockDim.x + threadIdx.x;
    if (i >= (long)B * 784) return;
    long b = i / 784; int j = (int)(i % 784);
    float s = 1.f / (1.f + expf(-c[b * C_W + j]));
    ENC[b * ENC_W + 784 + j] = x[i] - s;
}

__device__ __forceinline__ float gru_upd(float ir, float iz, float in_,
                                         float hr, float hz, float hn, float h) {
    float r = 1.f / (1.f + expf(-(ir + hr)));
    float z = 1.f / (1.f + expf(-(iz + hz)));
    float n = tanhf(in_ + r * hn);
    return (1.f - z) * n + z * h;
}

// GRU updates for h_mu and h_lv (in place in ENC)
__global__ void k_gate_enc(const float* __restrict__ GI,
                           const float* bih_mu, const float* bhh_mu,
                           const float* bih_lv, const float* bhh_lv,
                           float* __restrict__ ENC, int B) {
    long i = (long)blockIdx.x * blockDim.x + threadIdx.x;
    if (i >= (long)B * 100) return;
    long b = i / 100; int j = (int)(i % 100);
    const float* gi = GI + b * GI_W;
    {   // mu encoder
        float h = ENC[b * ENC_W + 1696 + j];
        float nh = gru_upd(gi[j] + bih_mu[j], gi[100 + j] + bih_mu[100 + j],
                           gi[200 + j] + bih_mu[200 + j],
                           gi[640 + j] + bhh_mu[j], gi[740 + j] + bhh_mu[100 + j],
                           gi[840 + j] + bhh_mu[200 + j], h);
        ENC[b * ENC_W + 1696 + j] = nh;
    }
    {   // lv encoder
        float h = ENC[b * ENC_W + 1824 + j];
        float nh = gru_upd(gi[320 + j] + bih_lv[j], gi[420 + j] + bih_lv[100 + j],
                           gi[520 + j] + bih_lv[200 + j],
                           gi[960 + j] + bhh_lv[j], gi[1060 + j] + bhh_lv[100 + j],
                           gi[1160 + j] + bhh_lv[200 + j], h);
        ENC[b * ENC_W + 1824 + j] = nh;
    }
}

// relu projections, reparameterize, emit mu/logvar, assemble ZH=[z|h_dec]
__global__ void k_reparam(const float* __restrict__ PRJ,
                          const float* bmu, const float* blv,
                          const float* __restrict__ noise_t,
                          const float* __restrict__ ENC,
                          float* __restrict__ ZH,
                          float* __restrict__ out_mu, float* __restrict__ out_lv,
                          int B) {
    long i = (long)blockIdx.x * blockDim.x + threadIdx.x;
    if (i >= (long)B * 100) return;
    long b = i / 100; int j = (int)(i % 100);
    float mu = fmaxf(PRJ[b * PRJ_W + j] + bmu[j], 0.f);
    float lv = fmaxf(PRJ[b * PRJ_W + 128 + j] + blv[j], 0.f);
    out_mu[i] = mu;
    out_lv[i] = lv;
    ZH[b * ZH_W + j]       = noise_t[i] * expf(0.5f * lv) + mu;
    ZH[b * ZH_W + 128 + j] = ENC[b * ENC_W + 1568 + j];   // h_dec^{t-1}
}

// decoder GRU update (h_dec in place in ENC)
__global__ void k_gate_dec(const float* __restrict__ GD,
                           const float* bih, const float* bhh,
                           float* __restrict__ ENC, int B) {
    long i = (long)blockIdx.x * blockDim.x + threadIdx.x;
    if (i >= (long)B * 100) return;
    long b = i / 100; int j = (int)(i % 100);
    const float* g = GD + b * GD_W;
    float h = ENC[b * ENC_W + 1568 + j];
    float nh = gru_upd(g[j] + bih[j], g[100 + j] + bih[100 + j], g[200 + j] + bih[200 + j],
                       g[320 + j] + bhh[j], g[420 + j] + bhh[100 + j], g[520 + j] + bhh[200 + j], h);
    ENC[b * ENC_W + 1568 + j] = nh;
}

__global__ void k_sigout(const float* __restrict__ c, float* __restrict__ out, int B) {
    long i = (long)blockIdx.x * blockDim.x + threadIdx.x;
    if (i >= (long)B * 784) return;
    long b = i / 784; int j = (int)(i % 784);
    out[i] = 1.f / (1.f + expf(-c[b * C_W + j]));
}

// ---------------------------------------------------------------------------
extern "C" void kernel_launch(void* const* d_in, const int* in_sizes, int n_in,
                              void* d_out, int out_size, void* d_ws, size_t ws_size,
                              hipStream_t stream) {
    const float* x_in    = (const float*)d_in[0];
    const float* noise   = (const float*)d_in[1];
    const float* Wih_mu  = (const float*)d_in[2];
    const float* Whh_mu  = (const float*)d_in[3];
    const float* bih_mu  = (const float*)d_in[4];
    const float* bhh_mu  = (const float*)d_in[5];
    const float* Wih_lv  = (const float*)d_in[6];
    const float* Whh_lv  = (const float*)d_in[7];
    const float* bih_lv  = (const float*)d_in[8];
    const float* bhh_lv  = (const float*)d_in[9];
    const float* Wmu     = (const float*)d_in[10];
    const float* bmu     = (const float*)d_in[11];
    const float* Wlv     = (const float*)d_in[12];
    const float* blv     = (const float*)d_in[13];
    const float* Wih_dec = (const float*)d_in[14];
    const float* Whh_dec = (const float*)d_in[15];
    const float* bih_dec = (const float*)d_in[16];
    const float* bhh_dec = (const float*)d_in[17];
    const float* Wwrite  = (const float*)d_in[18];
    const float* bwrite  = (const float*)d_in[19];

    const int B = in_sizes[0] / 784;                 // 8192 (multiple of 64)
    const int T = in_sizes[1] / (B * 100);           // 32

    // ---- carve workspace --------------------------------------------------
    float* ws = (float*)d_ws;
    size_t off = 0;
    auto carve = [&](size_t n) { float* p = ws + off; off += (n + 63) & ~(size_t)63; return p; };
    float* ENC  = carve((size_t)B * ENC_W);
    float* GI   = carve((size_t)B * GI_W);
    float* PRJ  = carve((size_t)B * PRJ_W);
    float* ZH   = carve((size_t)B * ZH_W);
    float* GD   = carve((size_t)B * GD_W);
    float* CBUF = carve((size_t)B * C_W);
    float* WENC = carve((size_t)ENC_W * GI_W);
    float* WPRJ = carve((size_t)256 * 256);
    float* WDEC = carve((size_t)256 * 640);
    float* WWR  = carve((size_t)128 * C_W);
    float* BWR  = carve((size_t)C_W);

    const int TPB = 256;
    auto blks = [&](long n) { return (unsigned)((n + TPB - 1) / TPB); };

    // ---- one-time (per call) setup ---------------------------------------
    init_enc<<<blks((long)B * ENC_W), TPB, 0, stream>>>(x_in, ENC, B);
    zero_f  <<<blks((long)B * C_W),   TPB, 0, stream>>>(CBUF, (long)B * C_W);
    zero_f  <<<blks((long)B * ZH_W),  TPB, 0, stream>>>(ZH,   (long)B * ZH_W);
    pack_wenc<<<blks((long)ENC_W * GI_W), TPB, 0, stream>>>(Wih_mu, Wih_lv, Whh_mu, Whh_lv, WENC);
    pack_wprj<<<blks(256 * 256), TPB, 0, stream>>>(Wmu, Wlv, WPRJ);
    pack_wdec<<<blks(256 * 640), TPB, 0, stream>>>(Wih_dec, Whh_dec, WDEC);
    pack_wwr <<<blks(128 * C_W), TPB, 0, stream>>>(Wwrite, WWR);
    pack_bwr <<<blks(C_W),       TPB, 0, stream>>>(bwrite, BWR);

    float* out_c  = (float*)d_out;
    float* out_mu = out_c + (size_t)B * 784;
    float* out_lv = out_mu + (size_t)T * B * 100;

    const unsigned Mt = (unsigned)(B / 64);

    for (int t = 0; t < T; ++t) {
        // x_hat from c^{t-1}
        k_xhat<<<blks((long)B * 784), TPB, 0, stream>>>(x_in, CBUF, ENC, B);

        // all four encoder gate blocks in one WMMA GEMM
        gemm_f32_wmma<<<dim3(GI_W / 64, Mt), 512, 0, stream>>>(
            ENC, ENC_W, WENC, GI_W, GI, GI_W, ENC_W, nullptr, 0);

        k_gate_enc<<<blks((long)B * 100), TPB, 0, stream>>>(
            GI, bih_mu, bhh_mu, bih_lv, bhh_lv, ENC, B);

        // mu/logvar projections (block-diagonal packed)
        gemm_f32_wmma<<<dim3(PRJ_W / 64, Mt), 512, 0, stream>>>(
            ENC + 1696, ENC_W, WPRJ, PRJ_W, PRJ, PRJ_W, 256, nullptr, 0);

        k_reparam<<<blks((long)B * 100), TPB, 0, stream>>>(
            PRJ, bmu, blv, noise + (size_t)t * B * 100, ENC, ZH,
            out_mu + (size_t)t * B * 100, out_lv + (size_t)t * B * 100, B);

        // decoder gi+gh in one GEMM
        gemm_f32_wmma<<<dim3(GD_W / 64, Mt), 512, 0, stream>>>(
            ZH, ZH_W, WDEC, GD_W, GD, GD_W, 256, nullptr, 0);

        k_gate_dec<<<blks((long)B * 100), TPB, 0, stream>>>(
            GD, bih_dec, bhh_dec, ENC, B);

        // canvas write: c += h_dec @ Wwrite^T + bwrite (accumulating GEMM)
        gemm_f32_wmma<<<dim3(C_W / 64, Mt), 512, 0, stream>>>(
            ENC + 1568, ENC_W, WWR, C_W, CBUF, C_W, 128, BWR, 1);
    }

    k_sigout<<<blks((long)B * 784), TPB, 0, stream>>>(CBUF, out_c, B);
}